// ResidualSeparableBlock_19756849562052
// MI455X (gfx1250) — compile-verified
//
#include <hip/hip_runtime.h>

typedef __attribute__((ext_vector_type(2))) float v2f;
typedef __attribute__((ext_vector_type(8))) float v8f;

#define NNODES 196608
#define BATCH  2
#define MROWS  (BATCH * NNODES)   // 393216 rows, divisible by 16
#define FDIM   64

// ---------------------------------------------------------------------------
// W_eff[k][f][o] = wd[k][f][0]*wp[2f][o] + wd[k][f][1]*wp[2f+1][o],
// stored PRE-PACKED in WMMA B-fragment order so each fragment is one b64:
//   f = 4*kk + 2*hi + c,  o = 16*ot + lo,  lane = hi*16 + lo
//   dst = k*4096 + (kk*4 + ot)*64 + lane*2 + c
// ---------------------------------------------------------------------------
__global__ __launch_bounds__(256) void build_weff(
    const float* __restrict__ wd, const float* __restrict__ wp,
    float* __restrict__ weff, int K)
{
  int idx = blockIdx.x * blockDim.x + threadIdx.x;
  int total = K * FDIM * FDIM;
  if (idx >= total) return;
  int o = idx & 63;
  int f = (idx >> 6) & 63;
  int k = idx >> 12;
  float w0 = wd[((k * FDIM) + f) * 2 + 0];
  float w1 = wd[((k * FDIM) + f) * 2 + 1];
  float val = w0 * wp[(2 * f) * FDIM + o] + w1 * wp[(2 * f + 1) * FDIM + o];

  int kk = f >> 2;
  int hi = (f >> 1) & 1;
  int c  = f & 1;
  int ot = o >> 4;
  int lo = o & 15;
  weff[k * 4096 + (kk * 4 + ot) * 64 + (hi * 16 + lo) * 2 + c] = val;
}

// ---------------------------------------------------------------------------
// Fused Chebyshev step + pointwise GEMM via V_WMMA_F32_16X16X4_F32.
// One wave32 per 16-row tile. MODE 0: T = t_in (no lap), acc = bias + T@W.
// MODE 1: T = lap(t_in), write t_out, acc += T@W.
// MODE 2: T = 2*lap(t_in) - t_sub, write t_out, acc += T@W.
//
// A-fragment (16x4 f32): lane l<16 -> row l, K = {0,1}; lane l+16 -> row l,
// K = {2,3}. Lane (lo,hi) holds, for each of 16 K-chunks kk, features
// f = 4*kk + 2*hi + {0,1} of row lo -> T is built directly in that layout.
// B-fragment: one aligned global_load_b64 from the pre-packed W (see above).
// C/D (16x16 f32): vgpr v, lane (lo,hi) -> row v+8*hi, col lo (per tile ot).
// ---------------------------------------------------------------------------
template <int MODE>
__global__ __launch_bounds__(256) void cheb_mm(
    const float* t_in, const float* t_sub, float* t_out, float* acc,
    const float* __restrict__ W, const float* __restrict__ bias,
    const int* __restrict__ nbr)
{
  const int lane = threadIdx.x & 31;
  const int wv   = threadIdx.x >> 5;
  const int lo   = lane & 15;
  const int hi   = lane >> 4;
  const long tile = (long)blockIdx.x * (blockDim.x >> 5) + wv;
  const long m0   = tile * 16;
  const long m    = m0 + lo;
  const int  n    = (int)(m % NNODES);
  const long rowbase = m * FDIM;

  v2f T[16];
  if (MODE == 0) {
#pragma unroll
    for (int kk = 0; kk < 16; ++kk)
      T[kk] = *(const v2f*)(t_in + rowbase + 4 * kk + 2 * hi);
  } else {
    int idx[8];
#pragma unroll
    for (int j = 0; j < 8; ++j) idx[j] = nbr[(long)n * 8 + j];
    const long bbase = (m / NNODES) * (long)NNODES * FDIM;
    v2f s[16];
#pragma unroll
    for (int kk = 0; kk < 16; ++kk) { s[kk].x = 0.0f; s[kk].y = 0.0f; }
#pragma unroll
    for (int j = 0; j < 8; ++j) {
      const float* prow = t_in + bbase + (long)idx[j] * FDIM + 2 * hi;
#pragma unroll
      for (int kk = 0; kk < 16; ++kk)
        s[kk] += *(const v2f*)(prow + 4 * kk);
    }
#pragma unroll
    for (int kk = 0; kk < 16; ++kk) {
      v2f own  = *(const v2f*)(t_in + rowbase + 4 * kk + 2 * hi);
      v2f lapv = own - s[kk] * 0.125f;   // v - mean(8 neighbors)
      if (MODE == 2) {
        v2f sub = *(const v2f*)(t_sub + rowbase + 4 * kk + 2 * hi);
        lapv = lapv * 2.0f - sub;        // Chebyshev recurrence
      }
      T[kk] = lapv;
      *(v2f*)(t_out + rowbase + 4 * kk + 2 * hi) = lapv;
    }
  }

  // Load / init the four 16x16 f32 accumulator fragments.
  v8f d[4];
  if (MODE == 0) {
#pragma unroll
    for (int ot = 0; ot < 4; ++ot) {
      float bv = bias[16 * ot + lo];
#pragma unroll
      for (int v = 0; v < 8; ++v) d[ot][v] = bv;
    }
  } else {
#pragma unroll
    for (int ot = 0; ot < 4; ++ot)
#pragma unroll
      for (int v = 0; v < 8; ++v)
        d[ot][v] = acc[(m0 + v + 8 * hi) * FDIM + 16 * ot + lo];
  }

  // 16 K-chunks x 4 output tiles = 64 WMMAs: (16x64) @ (64x64) in f32.
  // kk-outer so 4 consecutive WMMAs share the same A fragment.
#pragma unroll
  for (int kk = 0; kk < 16; ++kk) {
#pragma unroll
    for (int ot = 0; ot < 4; ++ot) {
      v2f bm = *(const v2f*)(W + (kk * 4 + ot) * 64 + lane * 2);
      d[ot] = __builtin_amdgcn_wmma_f32_16x16x4_f32(
          false, T[kk], false, bm, (short)0, d[ot], false, false);
    }
  }

#pragma unroll
  for (int ot = 0; ot < 4; ++ot)
#pragma unroll
    for (int v = 0; v < 8; ++v)
      acc[(m0 + v + 8 * hi) * FDIM + 16 * ot + lo] = d[ot][v];
}

// ---------------------------------------------------------------------------
// Per-node ReLU + LayerNorm (population var, /64) + optional residual.
// One wave32 per node row (64 feats = 2 per lane), shuffle reduction.
// ---------------------------------------------------------------------------
__global__ __launch_bounds__(256) void finalize_ln(
    float* x, const float* __restrict__ gamma, const float* __restrict__ beta,
    const float* residual)
{
  const int lane = threadIdx.x & 31;
  const int wv   = threadIdx.x >> 5;
  const long m   = (long)blockIdx.x * (blockDim.x >> 5) + wv;
  const long base = m * FDIM + lane * 2;

  v2f v = *(const v2f*)(x + base);
  v.x = fmaxf(v.x, 0.0f);
  v.y = fmaxf(v.y, 0.0f);

  float s = v.x + v.y;
#pragma unroll
  for (int off = 16; off >= 1; off >>= 1) s += __shfl_xor(s, off, 32);
  float mu = s * (1.0f / 64.0f);

  float dx = v.x - mu, dy = v.y - mu;
  float e = dx * dx + dy * dy;
#pragma unroll
  for (int off = 16; off >= 1; off >>= 1) e += __shfl_xor(e, off, 32);
  float rs = rsqrtf(e * (1.0f / 64.0f) + 1e-6f);

  v2f g = *(const v2f*)(gamma + lane * 2);
  v2f b = *(const v2f*)(beta + lane * 2);
  v2f o;
  o.x = dx * rs * g.x + b.x;
  o.y = dy * rs * g.y + b.y;
  if (residual) {
    v2f r = *(const v2f*)(residual + base);
    o.x += r.x;
    o.y += r.y;
  }
  *(v2f*)(x + base) = o;
}

// ---------------------------------------------------------------------------
extern "C" void kernel_launch(void* const* d_in, const int* in_sizes, int n_in,
                              void* d_out, int out_size, void* d_ws, size_t ws_size,
                              hipStream_t stream)
{
  const float* maps = (const float*)d_in[0];
  const int*   nbr  = (const int*)d_in[1];
  const float* wd1  = (const float*)d_in[2];
  const float* wp1  = (const float*)d_in[3];
  const float* b1   = (const float*)d_in[4];
  const float* g1   = (const float*)d_in[5];
  const float* bt1  = (const float*)d_in[6];
  const float* wd2  = (const float*)d_in[7];
  const float* wp2  = (const float*)d_in[8];
  const float* b2   = (const float*)d_in[9];
  const float* g2   = (const float*)d_in[10];
  const float* bt2  = (const float*)d_in[11];
  float* out = (float*)d_out;

  // Workspace layout (all 256B-aligned offsets).
  char*  ws    = (char*)d_ws;
  float* weff1 = (float*)(ws);                         //  4*64*64 f32 (packed)
  float* weff2 = (float*)(ws + 65536);                 //  8*64*64 f32 (packed)
  const size_t tb = (size_t)MROWS * FDIM * sizeof(float);  // ~100.7 MB
  float* bufA  = (float*)(ws + 196608);
  float* bufB  = (float*)(ws + 196608 + tb);
  float* x1    = (float*)(ws + 196608 + 2 * tb);
  (void)in_sizes; (void)n_in; (void)out_size; (void)ws_size;

  build_weff<<<(4 * 4096 + 255) / 256, 256, 0, stream>>>(wd1, wp1, weff1, 4);
  build_weff<<<(8 * 4096 + 255) / 256, 256, 0, stream>>>(wd2, wp2, weff2, 8);

  const int WPB   = 8;                  // waves (tiles) per 256-thread block
  const int tiles = MROWS / 16;         // 24576
  dim3 grid(tiles / WPB), blk(32 * WPB);

  // ----- Layer 1 (K = 4), accumulate into x1 -----
  cheb_mm<0><<<grid, blk, 0, stream>>>(maps, nullptr, nullptr, x1, weff1 + 0 * 4096, b1, nbr);
  cheb_mm<1><<<grid, blk, 0, stream>>>(maps, nullptr, bufA,    x1, weff1 + 1 * 4096, nullptr, nbr);
  cheb_mm<2><<<grid, blk, 0, stream>>>(bufA, maps,    bufB,    x1, weff1 + 2 * 4096, nullptr, nbr);
  cheb_mm<2><<<grid, blk, 0, stream>>>(bufB, bufA,    bufA,    x1, weff1 + 3 * 4096, nullptr, nbr);
  finalize_ln<<<MROWS / WPB, blk, 0, stream>>>(x1, g1, bt1, nullptr);

  // ----- Layer 2 (K = 8), accumulate into d_out -----
  cheb_mm<0><<<grid, blk, 0, stream>>>(x1,   nullptr, nullptr, out, weff2 + 0 * 4096, b2, nbr);
  cheb_mm<1><<<grid, blk, 0, stream>>>(x1,   nullptr, bufA,    out, weff2 + 1 * 4096, nullptr, nbr);
  cheb_mm<2><<<grid, blk, 0, stream>>>(bufA, x1,      bufB,    out, weff2 + 2 * 4096, nullptr, nbr);
  cheb_mm<2><<<grid, blk, 0, stream>>>(bufB, bufA,    bufA,    out, weff2 + 3 * 4096, nullptr, nbr);
  cheb_mm<2><<<grid, blk, 0, stream>>>(bufA, bufB,    bufB,    out, weff2 + 4 * 4096, nullptr, nbr);
  cheb_mm<2><<<grid, blk, 0, stream>>>(bufB, bufA,    bufA,    out, weff2 + 5 * 4096, nullptr, nbr);
  cheb_mm<2><<<grid, blk, 0, stream>>>(bufA, bufB,    bufB,    out, weff2 + 6 * 4096, nullptr, nbr);
  cheb_mm<2><<<grid, blk, 0, stream>>>(bufB, bufA,    bufA,    out, weff2 + 7 * 4096, nullptr, nbr);
  finalize_ln<<<MROWS / WPB, blk, 0, stream>>>(out, g2, bt2, maps);
}